// AUCMLoss_38147899523693
// MI455X (gfx1250) — compile-verified
//
#include <hip/hip_runtime.h>

typedef float v2f __attribute__((ext_vector_type(2)));
typedef float v8f __attribute__((ext_vector_type(8)));

#define NBLOCKS 1024
#define BLOCK 256
#define WAVES_PER_BLOCK (BLOCK / 32)

// ---------------------------------------------------------------------------
// Kernel 1: deterministic stream compaction of preds into pos[] / neg[]
// Single workgroup, LDS Hillis-Steele scan over per-thread chunk counts.
// ---------------------------------------------------------------------------
__global__ __launch_bounds__(1024) void aucm_compact(
    const float* __restrict__ preds, const int* __restrict__ targets, int n,
    float* __restrict__ posp, float* __restrict__ negp, int* __restrict__ counts) {
  __shared__ int cnt[1024];
  const int t = threadIdx.x;
  const int chunk = (n + 1023) >> 10;
  const int base = t * chunk;
  const int end = base + chunk < n ? base + chunk : n;

  int c = 0;
  for (int k = base; k < end; ++k) c += (targets[k] == 1) ? 1 : 0;
  cnt[t] = c;
  __syncthreads();
  // inclusive scan
  for (int off = 1; off < 1024; off <<= 1) {
    int add = (t >= off) ? cnt[t - off] : 0;
    __syncthreads();
    cnt[t] += add;
    __syncthreads();
  }
  int posStart = cnt[t] - c;        // exclusive prefix of positives
  int negStart = base - posStart;   // exclusive prefix of negatives
  int pi = posStart, ni = negStart;
  for (int k = base; k < end; ++k) {
    float v = preds[k];
    if (targets[k] == 1) posp[pi++] = v;
    else                 negp[ni++] = v;
  }
  if (t == 1023) {
    counts[0] = cnt[1023];       // P
    counts[1] = n - cnt[1023];   // M
  }
}

// ---------------------------------------------------------------------------
// Kernel 2: pairwise tile reduction.
// Each wave grid-strides over 16x16 (pos x neg) tiles. The tile of
// d_ij = 1 - p_i + n_j is produced by one V_WMMA_F32_16X16X4_F32 with
//   A row i = [-p_i, 1, 0, 0]   (16x4)
//   B col j = [ 1, 1+n_j, 0, 0] (4x16)
// then val = d*d + relu(d) (MARGIN = 1) is accumulated per lane.
// Uniform control flow per wave => EXEC stays all-ones around WMMA.
// ---------------------------------------------------------------------------
__global__ __launch_bounds__(BLOCK) void aucm_pairs(
    const float* __restrict__ posp, const float* __restrict__ negp,
    const int* __restrict__ counts, float* __restrict__ partials) {
  const int P = counts[0];
  const int M = counts[1];
  const int tilesP = (P + 15) >> 4;
  const int tilesM = (M + 15) >> 4;
  const long long ntiles = (long long)tilesP * (long long)tilesM;

  const int lane = threadIdx.x & 31;
  const int wave = threadIdx.x >> 5;
  const long long gw = (long long)blockIdx.x * WAVES_PER_BLOCK + wave;
  const long long gstride = (long long)gridDim.x * WAVES_PER_BLOCK;

  const int r = lane & 15;
  const float loMask = (lane < 16) ? 1.0f : 0.0f;  // lanes 0-15 carry K=0,1
  const int mbase = (lane >> 4) << 3;              // C/D rows: m = mbase + k
  const int Pm1 = (P > 0) ? P - 1 : 0;
  const int Mm1 = (M > 0) ? M - 1 : 0;

  float acc = 0.0f;

  for (long long tile = gw; tile < ntiles; tile += gstride) {
    const int ti = (int)(tile / tilesM);
    const int tj = (int)(tile - (long long)ti * tilesM);
    const int i0 = ti << 4;
    const int j0 = tj << 4;

    int ip = i0 + r; ip = ip < Pm1 ? ip : Pm1;   // clamped (masked later)
    int jn = j0 + r; jn = jn < Mm1 ? jn : Mm1;
    const float pv = posp[ip];
    const float nv = negp[jn];

    v2f a, b;
    a.x = -pv * loMask;          // A[m][0] = -p_m
    a.y = loMask;                // A[m][1] = 1
    b.x = loMask;                // B[0][n] = 1
    b.y = (1.0f + nv) * loMask;  // B[1][n] = 1 + n_n
    v8f c = {0.f, 0.f, 0.f, 0.f, 0.f, 0.f, 0.f, 0.f};
    c = __builtin_amdgcn_wmma_f32_16x16x4_f32(
        /*neg_a=*/false, a, /*neg_b=*/false, b,
        /*c_mod=*/(short)0, c, /*reuse_a=*/false, /*reuse_b=*/false);

    const bool interior = (i0 + 16 <= P) && (j0 + 16 <= M);  // uniform branch
    if (interior) {
#pragma unroll
      for (int k = 0; k < 8; ++k) {
        float d = c[k];
        acc += fmaf(d, d, fmaxf(d, 0.0f));
      }
    } else {
      const float colm = (j0 + r < M) ? 1.0f : 0.0f;
#pragma unroll
      for (int k = 0; k < 8; ++k) {
        float rowm = (i0 + mbase + k < P) ? 1.0f : 0.0f;
        float d = c[k];
        acc += fmaf(d, d, fmaxf(d, 0.0f)) * (rowm * colm);
      }
    }
  }

  // deterministic fixed-order block reduction
  __shared__ float red[BLOCK];
  red[threadIdx.x] = acc;
  __syncthreads();
  for (int off = BLOCK >> 1; off > 0; off >>= 1) {
    if (threadIdx.x < off) red[threadIdx.x] += red[threadIdx.x + off];
    __syncthreads();
  }
  if (threadIdx.x == 0) partials[blockIdx.x] = red[0];
}

// ---------------------------------------------------------------------------
// Kernel 3: fixed-order final sum in double, divide by P*M.
// ---------------------------------------------------------------------------
__global__ void aucm_finalize(const float* __restrict__ partials,
                              const int* __restrict__ counts,
                              float* __restrict__ out) {
  if (threadIdx.x == 0 && blockIdx.x == 0) {
    double s = 0.0;
    for (int i = 0; i < NBLOCKS; ++i) s += (double)partials[i];
    double np = (double)counts[0] * (double)counts[1];
    out[0] = (float)(s / np);
  }
}

extern "C" void kernel_launch(void* const* d_in, const int* in_sizes, int n_in,
                              void* d_out, int out_size, void* d_ws, size_t ws_size,
                              hipStream_t stream) {
  const float* preds = (const float*)d_in[0];
  const int* targets = (const int*)d_in[1];
  const int n = in_sizes[0];
  float* out = (float*)d_out;

  // workspace layout: [counts: 2 ints][pad][partials: NBLOCKS f32][posp: n f32][negp: n f32]
  int* counts = (int*)d_ws;
  float* partials = (float*)((char*)d_ws + 256);
  float* posp = (float*)((char*)d_ws + 256 + NBLOCKS * sizeof(float));
  float* negp = posp + n;

  aucm_compact<<<1, 1024, 0, stream>>>(preds, targets, n, posp, negp, counts);
  aucm_pairs<<<NBLOCKS, BLOCK, 0, stream>>>(posp, negp, counts, partials);
  aucm_finalize<<<1, 64, 0, stream>>>(partials, counts, out);
}